// NodeNetwork_43602507989842
// MI455X (gfx1250) — compile-verified
//
#include <hip/hip_runtime.h>

// ---------------------------------------------------------------------------
// GNN message-passing network on gfx1250 (MI455X).
// All dense GEMMs run on V_WMMA_F32_16X16X4_F32 (fp32 in/out, matches the
// fp32 reference). Edge gather/scatter (the bandwidth-dominant part,
// ~2.5 GB/call -> ~130us floor at 23.3 TB/s) uses coalesced float4 gathers +
// global_atomic_add_f32 scatters. Node tiles are staged through LDS.
// ---------------------------------------------------------------------------

typedef __attribute__((ext_vector_type(2))) float v2f;
typedef __attribute__((ext_vector_type(8))) float v8f;

#define LEAKY_SLOPE 0.01f
#define WAVES_PER_BLOCK 4

__device__ __forceinline__ int lane_id() { return threadIdx.x & 31; }

// A-fragment (16x4 f32): lanes 0-15 hold row M=lane, K = k0+{0,1};
// lanes 16-31 hold row M=lane-16, K = k0+{2,3}.  (ISA 7.12.2)
__device__ __forceinline__ v2f load_a_frag(const float* A, int lda, int k0) {
  int lane = lane_id();
  int row  = lane & 15;
  int koff = (lane >> 4) << 1;
  const float* p = A + row * lda + k0 + koff;
  v2f a; a.x = p[0]; a.y = p[1];
  return a;
}

// B-fragment (4x16 f32) from a row-major weight matrix W[K x ldw], cols n0..n0+15.
__device__ __forceinline__ v2f load_b_frag(const float* W, int ldw, int k0, int n0) {
  int lane = lane_id();
  int col  = lane & 15;
  int koff = (lane >> 4) << 1;
  v2f b;
  b.x = W[(k0 + koff)     * ldw + n0 + col];
  b.y = W[(k0 + koff + 1) * ldw + n0 + col];
  return b;
}

// B-fragment with column guard (for K x 24 weights padded to 2 col tiles).
__device__ __forceinline__ v2f load_b_frag_guard(const float* W, int ldw, int k0,
                                                 int n0, int ncols) {
  int lane = lane_id();
  int col  = n0 + (lane & 15);
  int koff = (lane >> 4) << 1;
  v2f b;
  if (col < ncols) {
    b.x = W[(k0 + koff)     * ldw + col];
    b.y = W[(k0 + koff + 1) * ldw + col];
  } else {
    b.x = 0.0f; b.y = 0.0f;
  }
  return b;
}

__device__ __forceinline__ v8f wmma_f32(v2f a, v2f b, v8f c) {
  // 8 args: (neg_a, A, neg_b, B, c_mod, C, reuse_a, reuse_b)
  return __builtin_amdgcn_wmma_f32_16x16x4_f32(false, a, false, b, (short)0, c,
                                               false, false);
}

// D/C tile (16x16 f32, 8 VGPRs): VGPR r holds row r (lanes 0-15) / r+8 (16-31).
__device__ __forceinline__ void store_d_tile(float* out, int ldo, int n0, v8f d) {
  int lane = lane_id();
  int col  = lane & 15;
  int hi   = lane >> 4;
#pragma unroll
  for (int r = 0; r < 8; ++r) out[(r + 8 * hi) * ldo + n0 + col] = d[r];
}

__device__ __forceinline__ v8f splat8(float v) {
  v8f a;
#pragma unroll
  for (int r = 0; r < 8; ++r) a[r] = v;
  return a;
}

__device__ __forceinline__ v8f leaky8(v8f a) {
#pragma unroll
  for (int r = 0; r < 8; ++r) {
    float v = a[r];
    a[r] = (v >= 0.0f) ? v : LEAKY_SLOPE * v;
  }
  return a;
}

// ---------------------------------------------------------------------------
// Encoder: h = leaky(x @ W1[16,128] + b1) @ W2[128,64] + b2.  Wave per 16 nodes.
// ---------------------------------------------------------------------------
__global__ __launch_bounds__(32 * WAVES_PER_BLOCK)
void encoder_kernel(const float* __restrict__ x,
                    const float* __restrict__ W1, const float* __restrict__ b1,
                    const float* __restrict__ W2, const float* __restrict__ b2,
                    float* __restrict__ h, int ntiles) {
  __shared__ float xs_[WAVES_PER_BLOCK][16 * 16];
  __shared__ float us_[WAVES_PER_BLOCK][16 * 128];
  int w    = threadIdx.x >> 5;
  int tile = blockIdx.x * WAVES_PER_BLOCK + w;
  bool active = tile < ntiles;
  int lane = lane_id();
  float* xS = xs_[w];
  float* uS = us_[w];

  if (active) {  // coalesced stage of the 16x16 x-tile
    const float4* g = (const float4*)(x + (size_t)tile * 16 * 16);
    float4* s4 = (float4*)xS;
    for (int i = lane; i < 64; i += 32) s4[i] = g[i];
  }
  __syncthreads();

  if (active) {
    int col = lane & 15;
#pragma unroll
    for (int ct = 0; ct < 8; ++ct) {          // 128 output cols
      int n0 = ct * 16;
      v8f acc = splat8(b1[n0 + col]);
#pragma unroll
      for (int k0 = 0; k0 < 16; k0 += 4)
        acc = wmma_f32(load_a_frag(xS, 16, k0), load_b_frag(W1, 128, k0, n0), acc);
      store_d_tile(uS, 128, n0, leaky8(acc));
    }
  }
  __syncthreads();

  if (active) {
    int col = lane & 15;
#pragma unroll
    for (int ct = 0; ct < 4; ++ct) {          // 64 output cols
      int n0 = ct * 16;
      v8f acc = splat8(b2[n0 + col]);
#pragma unroll
      for (int k0 = 0; k0 < 128; k0 += 4)
        acc = wmma_f32(load_a_frag(uS, 128, k0), load_b_frag(W2, 64, k0, n0), acc);
      store_d_tile(h + (size_t)tile * 16 * 64, 64, n0, acc);
    }
  }
}

// ---------------------------------------------------------------------------
// LayerNorm (jnp.var: 1/N) + zero the msg accumulator.  One wave per node.
// ---------------------------------------------------------------------------
__global__ __launch_bounds__(256)
void ln_zero_kernel(const float* __restrict__ h,
                    const float* __restrict__ g, const float* __restrict__ b,
                    float* __restrict__ hn, float* __restrict__ msg, int n) {
  int node = blockIdx.x * 8 + (threadIdx.x >> 5);
  if (node >= n) return;
  int lane = threadIdx.x & 31;
  float2 v = ((const float2*)(h + (size_t)node * 64))[lane];
  float s = v.x + v.y;
#pragma unroll
  for (int m = 16; m > 0; m >>= 1) s += __shfl_xor(s, m, 32);
  float mu = s * (1.0f / 64.0f);
  float dx = v.x - mu, dy = v.y - mu;
  float q = dx * dx + dy * dy;
#pragma unroll
  for (int m = 16; m > 0; m >>= 1) q += __shfl_xor(q, m, 32);
  float rstd = rsqrtf(q * (1.0f / 64.0f) + 1e-5f);
  float2 gg = ((const float2*)g)[lane];
  float2 bb = ((const float2*)b)[lane];
  float2 o;
  o.x = dx * rstd * gg.x + bb.x;
  o.y = dy * rstd * gg.y + bb.y;
  ((float2*)(hn + (size_t)node * 64))[lane] = o;
  float2 z; z.x = 0.0f; z.y = 0.0f;
  ((float2*)(msg + (size_t)node * 64))[lane] = z;
}

// ---------------------------------------------------------------------------
// Edge scatter: msg[dst] += w_e * hn[src].  Thread per (edge, 4-float chunk).
// ---------------------------------------------------------------------------
__global__ __launch_bounds__(256)
void scatter_kernel(const int* __restrict__ src, const int* __restrict__ dst,
                    const float* __restrict__ w, const float* __restrict__ hn,
                    float* __restrict__ msg, int nedges) {
  int t = blockIdx.x * 256 + threadIdx.x;
  int e = t >> 4;
  if (e >= nedges) return;
  int c = t & 15;
  int s = src[e], d = dst[e];
  float we = w[e];
  float4 v = ((const float4*)(hn + (size_t)s * 64))[c];
  float* mp = msg + (size_t)d * 64 + c * 4;
  atomicAdd(mp + 0, we * v.x);
  atomicAdd(mp + 1, we * v.y);
  atomicAdd(mp + 2, we * v.z);
  atomicAdd(mp + 3, we * v.w);
}

// ---------------------------------------------------------------------------
// Fused per-layer dense block:
//   y = hn@nW + nb + msg@eW + eb   (64->32)
//   h_out = leaky(y)@mW + mb + hn  (32->64, residual)
// Wave per 16-node tile; hn/msg/y staged in LDS.
// ---------------------------------------------------------------------------
__global__ __launch_bounds__(32 * WAVES_PER_BLOCK)
void layer_dense_kernel(const float* __restrict__ hn, const float* __restrict__ msg,
                        const float* __restrict__ nW, const float* __restrict__ nb,
                        const float* __restrict__ eW, const float* __restrict__ eb,
                        const float* __restrict__ mW, const float* __restrict__ mb,
                        float* __restrict__ hout, int ntiles) {
  __shared__ float hnS_[WAVES_PER_BLOCK][16 * 64];
  __shared__ float msgS_[WAVES_PER_BLOCK][16 * 64];
  __shared__ float yS_[WAVES_PER_BLOCK][16 * 32];
  int w    = threadIdx.x >> 5;
  int tile = blockIdx.x * WAVES_PER_BLOCK + w;
  bool active = tile < ntiles;
  int lane = lane_id();
  float* hnS  = hnS_[w];
  float* msgS = msgS_[w];
  float* yS   = yS_[w];

  if (active) {  // coalesced stage: 16x64 hn tile + 16x64 msg tile
    const float4* gh = (const float4*)(hn  + (size_t)tile * 16 * 64);
    const float4* gm = (const float4*)(msg + (size_t)tile * 16 * 64);
    float4* sh = (float4*)hnS;
    float4* sm = (float4*)msgS;
    for (int i = lane; i < 256; i += 32) { sh[i] = gh[i]; sm[i] = gm[i]; }
  }
  __syncthreads();

  if (active) {
    int col = lane & 15;
#pragma unroll
    for (int ct = 0; ct < 2; ++ct) {          // 32 output cols
      int n0 = ct * 16;
      v8f acc = splat8(nb[n0 + col] + eb[n0 + col]);
#pragma unroll
      for (int k0 = 0; k0 < 64; k0 += 4) {
        acc = wmma_f32(load_a_frag(hnS, 64, k0),  load_b_frag(nW, 32, k0, n0), acc);
        acc = wmma_f32(load_a_frag(msgS, 64, k0), load_b_frag(eW, 32, k0, n0), acc);
      }
      store_d_tile(yS, 32, n0, leaky8(acc));  // act_first
    }
  }
  __syncthreads();

  if (active) {
    int col = lane & 15;
    int hi  = lane >> 4;
#pragma unroll
    for (int ct = 0; ct < 4; ++ct) {          // 64 output cols
      int n0 = ct * 16;
      v8f acc = splat8(mb[n0 + col]);
#pragma unroll
      for (int k0 = 0; k0 < 32; k0 += 4)
        acc = wmma_f32(load_a_frag(yS, 32, k0), load_b_frag(mW, 64, k0, n0), acc);
#pragma unroll
      for (int r = 0; r < 8; ++r)             // residual: + hn
        acc[r] += hnS[(r + 8 * hi) * 64 + n0 + col];
      store_d_tile(hout + (size_t)tile * 16 * 64, 64, n0, acc);
    }
  }
}

// ---------------------------------------------------------------------------
// Decoder: out = leaky(h @ W1[64,24] + b1) @ W2[24,3] + b2.  Wave per 16 nodes.
// ---------------------------------------------------------------------------
__global__ __launch_bounds__(32 * WAVES_PER_BLOCK)
void decoder_kernel(const float* __restrict__ h,
                    const float* __restrict__ W1, const float* __restrict__ b1,
                    const float* __restrict__ W2, const float* __restrict__ b2,
                    float* __restrict__ out, int ntiles) {
  __shared__ float hS_[WAVES_PER_BLOCK][16 * 64];
  __shared__ float tS_[WAVES_PER_BLOCK][16 * 24];
  int w    = threadIdx.x >> 5;
  int tile = blockIdx.x * WAVES_PER_BLOCK + w;
  bool active = tile < ntiles;
  int lane = lane_id();
  float* hS = hS_[w];
  float* tS = tS_[w];

  if (active) {
    const float4* g = (const float4*)(h + (size_t)tile * 16 * 64);
    float4* s4 = (float4*)hS;
    for (int i = lane; i < 256; i += 32) s4[i] = g[i];
  }
  __syncthreads();

  if (active) {
    int col = lane & 15;
    int hi  = lane >> 4;
#pragma unroll
    for (int ct = 0; ct < 2; ++ct) {          // 24 cols padded to 2 tiles
      int n0 = ct * 16;
      int c  = n0 + col;
      v8f acc = splat8((c < 24) ? b1[c] : 0.0f);
#pragma unroll
      for (int k0 = 0; k0 < 64; k0 += 4)
        acc = wmma_f32(load_a_frag(hS, 64, k0),
                       load_b_frag_guard(W1, 24, k0, n0, 24), acc);
      if (c < 24) {
#pragma unroll
        for (int r = 0; r < 8; ++r) {
          float v = acc[r];
          v = (v >= 0.0f) ? v : LEAKY_SLOPE * v;
          tS[(r + 8 * hi) * 24 + c] = v;
        }
      }
    }
  }
  __syncthreads();

  if (active && lane < 16) {                  // tiny 24->3 tail, scalar per node
    int node = tile * 16 + lane;
    float r0 = b2[0], r1 = b2[1], r2 = b2[2];
#pragma unroll
    for (int k = 0; k < 24; ++k) {
      float tv = tS[lane * 24 + k];
      r0 += tv * W2[k * 3 + 0];
      r1 += tv * W2[k * 3 + 1];
      r2 += tv * W2[k * 3 + 2];
    }
    out[node * 3 + 0] = r0;
    out[node * 3 + 1] = r1;
    out[node * 3 + 2] = r2;
  }
}

// ---------------------------------------------------------------------------
extern "C" void kernel_launch(void* const* d_in, const int* in_sizes, int n_in,
                              void* d_out, int out_size, void* d_ws, size_t ws_size,
                              hipStream_t stream) {
  (void)n_in; (void)out_size; (void)ws_size;
  const float* x      = (const float*)d_in[0];
  /* d_in[1] = pos: unused by the reference */
  const int*   esrc   = (const int*)d_in[2];
  const int*   edst   = (const int*)d_in[3];
  const float* ew     = (const float*)d_in[4];
  const float* encW1  = (const float*)d_in[5];
  const float* encb1  = (const float*)d_in[6];
  const float* encW2  = (const float*)d_in[7];
  const float* encb2  = (const float*)d_in[8];
  const float* decW1  = (const float*)d_in[9];
  const float* decb1  = (const float*)d_in[10];
  const float* decW2  = (const float*)d_in[11];
  const float* decb2  = (const float*)d_in[12];
  const float* ln_g   = (const float*)d_in[13];
  const float* ln_b   = (const float*)d_in[14];
  const float* nodeW  = (const float*)d_in[15];
  const float* nodeB  = (const float*)d_in[16];
  const float* edgeW  = (const float*)d_in[17];
  const float* edgeB  = (const float*)d_in[18];
  const float* mlpW   = (const float*)d_in[19];
  const float* mlpB   = (const float*)d_in[20];

  const int N = in_sizes[0] / 16;   // 100000
  const int E = in_sizes[2];        // 1200000
  const int ntiles = (N + 15) / 16; // N is a multiple of 16 (6250 tiles)
  const int tblocks = (ntiles + WAVES_PER_BLOCK - 1) / WAVES_PER_BLOCK;

  // Workspace layout: h ping-pong + hn + msg (4 * N*64 fp32 ~= 102 MB)
  float* h0  = (float*)d_ws;
  float* h1  = h0 + (size_t)N * 64;
  float* hn  = h1 + (size_t)N * 64;
  float* msg = hn + (size_t)N * 64;

  encoder_kernel<<<tblocks, 32 * WAVES_PER_BLOCK, 0, stream>>>(
      x, encW1, encb1, encW2, encb2, h0, ntiles);

  float* cur = h0;
  float* nxt = h1;
  for (int l = 0; l < 4; ++l) {
    ln_zero_kernel<<<(N + 7) / 8, 256, 0, stream>>>(
        cur, ln_g + l * 64, ln_b + l * 64, hn, msg, N);
    int sthreads_blocks = (E * 16 + 255) / 256;
    scatter_kernel<<<sthreads_blocks, 256, 0, stream>>>(esrc, edst, ew, hn, msg, E);
    layer_dense_kernel<<<tblocks, 32 * WAVES_PER_BLOCK, 0, stream>>>(
        hn, msg,
        nodeW + (size_t)l * 64 * 32, nodeB + (size_t)l * 32,
        edgeW + (size_t)l * 64 * 32, edgeB + (size_t)l * 32,
        mlpW  + (size_t)l * 32 * 64, mlpB  + (size_t)l * 64,
        nxt, ntiles);
    float* t = cur; cur = nxt; nxt = t;
  }

  decoder_kernel<<<tblocks, 32 * WAVES_PER_BLOCK, 0, stream>>>(
      cur, decW1, decb1, decW2, decb2, (float*)d_out, ntiles);
}